// RGBDModule_36807869726995
// MI455X (gfx1250) — compile-verified
//
#include <hip/hip_runtime.h>
#include <math.h>

// ---------------------------------------------------------------------------
// Problem constants (match the reference)
// ---------------------------------------------------------------------------
#define HH     1280
#define WW     1920
#define OH     427
#define OW     640
#define NB     2
#define NPTS   120000
#define PLANE  (HH * WW)          // 2,457,600 pixels per batch plane
#define TOTAL  (NB * PLANE)       // 4,915,200
#define INVALID 0xFFFFFFFFu

typedef __attribute__((ext_vector_type(2))) float v2f;
typedef __attribute__((ext_vector_type(8))) float v8f;

#if defined(__has_builtin)
#  if __has_builtin(__builtin_amdgcn_global_load_async_to_lds_b128)
#    define USE_ASYNC_LDS 1
// Exact parameter types per hipcc diagnostic: param0 = global (AS1) v4i*,
// non-const; param1 = LDS (AS3) v4i*.
typedef int v4i __attribute__((__vector_size__(4 * sizeof(int))));
typedef __attribute__((address_space(1))) v4i* gv4i_p;
typedef __attribute__((address_space(3))) v4i* lv4i_p;
#  endif
#endif

// ---------------------------------------------------------------------------
// Zero the depth canvas (b128 stores)
// ---------------------------------------------------------------------------
__global__ void k_zero4(float* __restrict__ p) {
    int t = blockIdx.x * blockDim.x + threadIdx.x;
    if (t < TOTAL / 4) {
        float4 z = make_float4(0.f, 0.f, 0.f, 0.f);
        *(float4*)(p + t * 4) = z;
    }
}

// ---------------------------------------------------------------------------
// Projection + scatter.  One wave handles a tile of 16 points via
// V_WMMA_F32_16X16X4_F32:  D(16x16) = A(16x4 homogeneous points) x
// B(4x16, columns 0..2 = rows of the fused 3x4 matrix M = K*E*P | t).
// ---------------------------------------------------------------------------
__global__ void k_proj(const float* __restrict__ pts,
                       const float* __restrict__ pose,
                       const float* __restrict__ extr,
                       const float* __restrict__ intr,
                       float* __restrict__ depth) {
    __shared__ float lds[8][16][4];
    const int lane   = threadIdx.x & 31;
    const int wv     = threadIdx.x >> 5;
    const int waveId = blockIdx.x * (blockDim.x >> 5) + wv;
    const int numTiles = (NB * NPTS) / 16;          // 15000
    if (waveId >= numTiles) return;                 // uniform per wave -> EXEC full

    const int tile = waveId;
    const int b    = (tile * 16) / NPTS;

    float M[3][4];
    {
        const float* P = pose + b * 16;
        const float* E = extr + b * 16;
        const float* K = intr + b * 9;
        float R[3][3], tt[3];
        for (int i = 0; i < 3; ++i) {
            for (int j = 0; j < 3; ++j) {
                float s = 0.f;
                for (int k = 0; k < 3; ++k) s += E[i * 4 + k] * P[k * 4 + j];
                R[i][j] = s;
            }
            float s = 0.f;
            for (int k = 0; k < 3; ++k) s += E[i * 4 + k] * P[k * 4 + 3];
            tt[i] = s + E[i * 4 + 3];
        }
        for (int i = 0; i < 3; ++i) {
            for (int j = 0; j < 3; ++j) {
                float s = 0.f;
                for (int k = 0; k < 3; ++k) s += K[i * 3 + k] * R[k][j];
                M[i][j] = s;
            }
            float s = 0.f;
            for (int k = 0; k < 3; ++k) s += K[i * 3 + k] * tt[k];
            M[i][3] = s;
        }
    }

    const int m = lane & 15;
    const int q = tile * 16 + m;
    const float px = pts[q * 3 + 0];
    const float py = pts[q * 3 + 1];
    const float pz = pts[q * 3 + 2];

    // A (16x4 f32): lanes 0-15 hold K=0,1 ; lanes 16-31 hold K=2,3
    v2f a;
    if (lane < 16) { a.x = px;  a.y = py;   }
    else           { a.x = pz;  a.y = 1.0f; }

    // B (4x16 f32): VGPR v, lanes 0-15 -> K=v, lanes 16-31 -> K=v+2; N = lane&15
    const int n = lane & 15;
    v2f bm;
    if (n < 3) {
        const int k0 = (lane < 16) ? 0 : 2;
        bm.x = M[n][k0];
        bm.y = M[n][k0 + 1];
    } else {
        bm.x = 0.0f; bm.y = 0.0f;
    }

    v8f c = {};
    v8f d = __builtin_amdgcn_wmma_f32_16x16x4_f32(
        false, a, false, bm, (short)0, c, false, false);

    if (n < 3) {
        const int rbase = (lane < 16) ? 0 : 8;
        #pragma unroll
        for (int r = 0; r < 8; ++r)
            lds[wv][rbase + r][n] = d[r];
    }
    asm volatile("s_wait_dscnt 0" ::: "memory");

    if (lane < 16) {
        const float U = lds[wv][m][0];
        const float V = lds[wv][m][1];
        const float Z = lds[wv][m][2];
        const float u = U / Z;
        const float v = V / Z;
        const int r  = (int)floorf(v);
        const int cc = (int)floorf(u);
        if (r >= 0 && r < HH && cc >= 0 && cc < WW) {
            const float dep = sqrtf(px * px + py * py + pz * pz);
            atomicAdd(&depth[b * PLANE + r * WW + cc], dep);
        }
    }
}

// ---------------------------------------------------------------------------
// JFA seed init, 4 pixels per thread (b128 in / b128 out)
// ---------------------------------------------------------------------------
__global__ void k_jfa_init4(const float* __restrict__ depth, unsigned* __restrict__ dst) {
    int t = blockIdx.x * blockDim.x + threadIdx.x;
    if (t >= TOTAL / 4) return;
    int i0 = t * 4;
    int x0 = i0 % WW;
    int y  = (i0 / WW) % HH;
    float4 d = *(const float4*)(depth + i0);
    uint4 o;
    o.x = (d.x != 0.0f) ? (unsigned)((y << 16) | (x0 + 0)) : INVALID;
    o.y = (d.y != 0.0f) ? (unsigned)((y << 16) | (x0 + 1)) : INVALID;
    o.z = (d.z != 0.0f) ? (unsigned)((y << 16) | (x0 + 2)) : INVALID;
    o.w = (d.w != 0.0f) ? (unsigned)((y << 16) | (x0 + 3)) : INVALID;
    *(uint4*)(dst + i0) = o;
}

__device__ __forceinline__ unsigned seed_d2(unsigned s, int y, int x) {
    if (s == INVALID) return INVALID;                 // > any real d^2 (< 7.37e6)
    int dr = y - (int)(s >> 16);
    int dc = x - (int)(s & 0xFFFFu);
    return (unsigned)(dr * dr + dc * dc);             // exact; matches reference f32 math
}

// ---------------------------------------------------------------------------
// Scalar JFA step (used for k = 1, 2). Loop order matches the reference so
// strict-< tie-breaking is identical.
// ---------------------------------------------------------------------------
__global__ void k_jfa_step(const unsigned* __restrict__ src, unsigned* __restrict__ dst, int k) {
    int i = blockIdx.x * blockDim.x + threadIdx.x;
    if (i >= TOTAL) return;
    int x = i % WW;
    int y = (i / WW) % HH;
    int b = i / PLANE;
    const unsigned* pl = src + b * PLANE;

    unsigned best = pl[y * WW + x];
    unsigned bd   = seed_d2(best, y, x);

    #pragma unroll
    for (int ai = 0; ai < 3; ++ai) {
        int dy = (ai == 0) ? -k : (ai == 1 ? 0 : k);
        #pragma unroll
        for (int bi = 0; bi < 3; ++bi) {
            int dx = (bi == 0) ? -k : (bi == 1 ? 0 : k);
            if (dy == 0 && dx == 0) continue;
            int sy = y - dy, sx = x - dx;
            if (sy < 0 || sy >= HH || sx < 0 || sx >= WW) continue;
            unsigned cand = pl[sy * WW + sx];
            unsigned dd   = seed_d2(cand, y, x);
            if (dd < bd) { bd = dd; best = cand; }
        }
    }
    dst[i] = best;
}

// ---------------------------------------------------------------------------
// Vectorized JFA step for k % 4 == 0 (radii 1024..4): 4 pixels per thread,
// 9 b128 loads + 1 b128 store per 4 pixels (4x fewer VMEM issues).
// Chunk-level bounds checks are exactly the per-pixel ones because
// x0, k, W are all multiples of 4.
// ---------------------------------------------------------------------------
__global__ void k_jfa_step4(const unsigned* __restrict__ src, unsigned* __restrict__ dst, int k) {
    int t = blockIdx.x * blockDim.x + threadIdx.x;
    if (t >= TOTAL / 4) return;
    int i0 = t * 4;
    int x0 = i0 % WW;
    int y  = (i0 / WW) % HH;
    int b  = i0 / PLANE;
    const unsigned* pl = src + b * PLANE;

    uint4 cur = *(const uint4*)(pl + y * WW + x0);
    unsigned be[4] = {cur.x, cur.y, cur.z, cur.w};
    unsigned bd[4];
    #pragma unroll
    for (int j = 0; j < 4; ++j) bd[j] = seed_d2(be[j], y, x0 + j);

    #pragma unroll
    for (int ai = 0; ai < 3; ++ai) {
        int dy = (ai == 0) ? -k : (ai == 1 ? 0 : k);
        int sy = y - dy;
        if (sy < 0 || sy >= HH) continue;
        const unsigned* row = pl + sy * WW;
        #pragma unroll
        for (int bi = 0; bi < 3; ++bi) {
            int dx = (bi == 0) ? -k : (bi == 1 ? 0 : k);
            if (dy == 0 && dx == 0) continue;
            int cb = x0 - dx;                       // aligned (x0%4==0, k%4==0)
            if (cb < 0 || cb + 4 > WW) continue;    // all-in or all-out
            uint4 ck = *(const uint4*)(row + cb);
            unsigned ce[4] = {ck.x, ck.y, ck.z, ck.w};
            #pragma unroll
            for (int j = 0; j < 4; ++j) {
                unsigned dd = seed_d2(ce[j], y, x0 + j);
                if (dd < bd[j]) { bd[j] = dd; be[j] = ce[j]; }
            }
        }
    }
    uint4 o; o.x = be[0]; o.y = be[1]; o.z = be[2]; o.w = be[3];
    *(uint4*)(dst + i0) = o;
}

// ---------------------------------------------------------------------------
// Fill + distance. depth is filled IN PLACE: writes touch only invalid pixels,
// reads touch only valid (seed) pixels -> race free.
// ---------------------------------------------------------------------------
__global__ void k_finalize(float* __restrict__ depth,
                           const unsigned* __restrict__ nearest,
                           float* __restrict__ dist) {
    int i = blockIdx.x * blockDim.x + threadIdx.x;
    if (i >= TOTAL) return;
    int x = i % WW;
    int y = (i / WW) % HH;
    int b = i / PLANE;
    float dv = depth[i];
    if (dv != 0.0f) {
        dist[i] = 0.0f;
    } else {
        unsigned s = nearest[i];
        int sr = (int)(s >> 16);     if (sr > HH - 1) sr = HH - 1;   // matches jnp.clip
        int sc = (int)(s & 0xFFFFu); if (sc > WW - 1) sc = WW - 1;
        float dr = (float)(y - sr), dc = (float)(x - sc);
        dist[i]  = sqrtf(dr * dr + dc * dc);
        depth[i] = depth[b * PLANE + sr * WW + sc];
    }
}

// ---------------------------------------------------------------------------
// Resize: jax.image.resize(method='linear', antialias=True) downscale.
// One block per (b, ch, oy) output row. Each input pixel is reused ~4x across
// output taps, so the <=8 needed input rows are staged into LDS — with the
// CDNA5 async global->LDS copy path when the builtin is available.
// ---------------------------------------------------------------------------
__global__ void k_resize(const float* __restrict__ filled,
                         const float* __restrict__ dist,
                         float* __restrict__ out) {
    __shared__ float shm[8 * WW];                    // 61.4 KB of the 320 KB LDS

    const int oy = blockIdx.x % OH;
    const int ch = (blockIdx.x / OH) % 2;
    const int b  = blockIdx.x / (OH * 2);
    const float* src = (ch == 0 ? filled : dist) + b * PLANE;

    const float sy = (float)OH / (float)HH;          // antialias kernel scale (<1)
    const float sx = (float)OW / (float)WW;
    const float ry = (float)HH / (float)OH;          // support radius = 1/scale
    const float rx = (float)WW / (float)OW;
    const float cy = (oy + 0.5f) * ry - 0.5f;

    int y0 = (int)floorf(cy - ry); if (y0 < 0) y0 = 0;
    int y1 = (int)ceilf (cy + ry); if (y1 > HH - 1) y1 = HH - 1;     // y1-y0+1 <= 8

    // ---- stage rows [y0, y1] into LDS (uniform bounds -> barrier-safe) ----
    const int nrows  = y1 - y0 + 1;
    const int chunks = nrows * (WW / 4);
    for (int c = threadIdx.x; c < chunks; c += blockDim.x) {
        int rr = c / (WW / 4);
        int c4 = (c % (WW / 4)) * 4;
        const float* g = src + (y0 + rr) * WW + c4;
        float* l = &shm[rr * WW + c4];
#if defined(USE_ASYNC_LDS)
        __builtin_amdgcn_global_load_async_to_lds_b128(
            (gv4i_p)g, (lv4i_p)l, 0, 0);
#else
        *(float4*)l = *(const float4*)g;
#endif
    }
#if defined(USE_ASYNC_LDS)
    asm volatile("s_wait_asynccnt 0" ::: "memory");
#endif
    __syncthreads();

    // ---- per-output-pixel separable triangle filter from LDS ----
    float wys = 0.0f;
    float wy[8];
    #pragma unroll
    for (int rr = 0; rr < 8; ++rr) {
        float w = 0.0f;
        if (rr < nrows) {
            w = 1.0f - fabsf((float)(y0 + rr) - cy) * sy;
            if (w < 0.0f) w = 0.0f;
        }
        wy[rr] = w;
        wys += w;
    }

    for (int ox = threadIdx.x; ox < OW; ox += blockDim.x) {
        const float cx = (ox + 0.5f) * rx - 0.5f;
        int x0 = (int)floorf(cx - rx); if (x0 < 0) x0 = 0;
        int x1 = (int)ceilf (cx + rx); if (x1 > WW - 1) x1 = WW - 1;

        float wxs = 0.0f, acc = 0.0f;
        for (int ix = x0; ix <= x1; ++ix) {
            float wx = 1.0f - fabsf((float)ix - cx) * sx;
            if (wx <= 0.0f) continue;
            wxs += wx;
            float col = 0.0f;
            #pragma unroll
            for (int rr = 0; rr < 8; ++rr)
                if (rr < nrows) col += wy[rr] * shm[rr * WW + ix];
            acc += wx * col;
        }
        out[((b * 2 + ch) * OH + oy) * OW + ox] = acc / (wys * wxs);
    }
}

// ---------------------------------------------------------------------------
// Launcher.  Workspace: depth (TOTAL f32) | bufA (TOTAL u32) | bufB (TOTAL u32)
// = 59 MB (fully L2-resident on MI455X's 192 MB L2); bufB is recycled as the
// distance plane after the 12 (even) JFA steps leave the final seeds in bufA.
// ---------------------------------------------------------------------------
extern "C" void kernel_launch(void* const* d_in, const int* in_sizes, int n_in,
                              void* d_out, int out_size, void* d_ws, size_t ws_size,
                              hipStream_t stream) {
    const float* pts  = (const float*)d_in[0];   // (B, N, 3)
    const float* pose = (const float*)d_in[1];   // (B, 4, 4)
    const float* extr = (const float*)d_in[2];   // (B, 4, 4)
    const float* intr = (const float*)d_in[3];   // (B, 3, 3)
    float* out = (float*)d_out;                  // (B, 2, OH, OW)

    float*    depth = (float*)d_ws;
    unsigned* bufA  = (unsigned*)(depth + TOTAL);
    unsigned* bufB  = bufA + TOTAL;
    float*    dist  = (float*)bufB;              // reused once JFA is done

    const int T     = 256;
    const int gPix  = (TOTAL + T - 1) / T;
    const int gPix4 = (TOTAL / 4 + T - 1) / T;

    k_zero4<<<gPix4, T, 0, stream>>>(depth);

    const int numTiles = (NB * NPTS) / 16;       // 15000 waves, 8 waves/block
    k_proj<<<(numTiles + 7) / 8, 256, 0, stream>>>(pts, pose, extr, intr, depth);

    k_jfa_init4<<<gPix4, T, 0, stream>>>(depth, bufA);

    const int steps[12] = {1, 1024, 512, 256, 128, 64, 32, 16, 8, 4, 2, 1};
    unsigned* cur = bufA;
    unsigned* nxt = bufB;
    for (int s = 0; s < 12; ++s) {
        if ((steps[s] & 3) == 0)
            k_jfa_step4<<<gPix4, T, 0, stream>>>(cur, nxt, steps[s]);
        else
            k_jfa_step<<<gPix, T, 0, stream>>>(cur, nxt, steps[s]);
        unsigned* t = cur; cur = nxt; nxt = t;
    }
    // 12 swaps -> cur == bufA, bufB free for dist

    k_finalize<<<gPix, T, 0, stream>>>(depth, cur, dist);

    k_resize<<<NB * 2 * OH, T, 0, stream>>>(depth, dist, out);
}